// AffinityPropagate_65798898974991
// MI455X (gfx1250) — compile-verified
//
#include <hip/hip_runtime.h>

#define HH   544
#define WW   960
#define BB   8
#define HWsz (HH * WW)
#define TX   32
#define TY   8

// One propagation iteration:
// out[b,y,x] = (sum_k |aff[b,k,y,x]| * in[b, y+dy, x+dx]) / (sum_k |aff[b,k,y,x]|)
// with zero padding at the image border.
__global__ __launch_bounds__(TX * TY)
void ap_step(const float* __restrict__ aff,
             const float* __restrict__ fin,
             float* __restrict__ fout) {
    // (TY+2) x (TX+2) feature halo tile in LDS
    __shared__ float tile[(TY + 2) * (TX + 2)];

    const int tx  = threadIdx.x;
    const int ty  = threadIdx.y;
    const int gx0 = blockIdx.x * TX;
    const int gy0 = blockIdx.y * TY;
    const int b   = blockIdx.z;

    const float* fb = fin + (size_t)b * HWsz;

    // Cooperative halo fill via CDNA5 async global->LDS DMA (no VGPR staging).
    // Out-of-bounds entries are zero-filled through the normal DS path.
    for (int i = ty * TX + tx; i < (TY + 2) * (TX + 2); i += TX * TY) {
        const int ly = i / (TX + 2);
        const int lx = i % (TX + 2);
        const int gy = gy0 + ly - 1;
        const int gx = gx0 + lx - 1;
        if (gy >= 0 && gy < HH && gx >= 0 && gx < WW) {
            // Low 32 bits of a generic (flat) shared-memory address are the
            // LDS byte offset (ISA 10.2).
            unsigned lds_addr = (unsigned)(size_t)&tile[i];
            const float* gp = fb + (size_t)gy * WW + gx;
            asm volatile("global_load_async_to_lds_b32 %0, %1, off"
                         :: "v"(lds_addr), "v"(gp)
                         : "memory");
        } else {
            tile[i] = 0.0f;   // ds_store, covered by __syncthreads' dscnt wait
        }
    }
    // Wait for this wave's async DMAs, then make LDS visible workgroup-wide.
    asm volatile("s_wait_asynccnt 0x0" ::: "memory");
    __syncthreads();

    const int gy = gy0 + ty;
    const int gx = gx0 + tx;
    const float* ap = aff + (size_t)b * 9 * HWsz + (size_t)gy * WW + gx;

    float num = 0.0f;
    float den = 0.0f;
#pragma unroll
    for (int k = 0; k < 9; ++k) {
        const int dy = k / 3;
        const int dx = k % 3;
        const float a = __builtin_fabsf(ap[(size_t)k * HWsz]);  // coalesced 128B/plane
        const float f = tile[(ty + dy) * (TX + 2) + (tx + dx)];
        num = __builtin_fmaf(a, f, num);
        den += a;
    }
    fout[(size_t)b * HWsz + (size_t)gy * WW + gx] = num / den;
}

extern "C" void kernel_launch(void* const* d_in, const int* in_sizes, int n_in,
                              void* d_out, int out_size, void* d_ws, size_t ws_size,
                              hipStream_t stream) {
    (void)in_sizes; (void)n_in; (void)out_size; (void)ws_size;

    const float* aff  = (const float*)d_in[0];   // [8, 9, 544, 960] f32
    const float* feat = (const float*)d_in[1];   // [8, 1, 544, 960] f32
    float*       out  = (float*)d_out;           // [8, 1, 544, 960] f32
    float*       ws   = (float*)d_ws;            // >= 16.7 MB scratch

    dim3 grid(WW / TX, HH / TY, BB);   // 30 x 68 x 8, exact tiling
    dim3 block(TX, TY, 1);             // 256 threads = 8 wave32s

    // 4 fused-normalization iterations, ping-ponging buffers; the 150 MB
    // affinity tensor stays L2-resident (192 MB L2) across all launches.
    ap_step<<<grid, block, 0, stream>>>(aff, feat, ws);   // it 0: feat -> ws
    ap_step<<<grid, block, 0, stream>>>(aff, ws,   out);  // it 1: ws   -> out
    ap_step<<<grid, block, 0, stream>>>(aff, out,  ws);   // it 2: out  -> ws
    ap_step<<<grid, block, 0, stream>>>(aff, ws,   out);  // it 3: ws   -> out
}